// Theta_88579405513273
// MI455X (gfx1250) — compile-verified
//
#include <hip/hip_runtime.h>
#include <hip/hip_bf16.h>

// ---------------------------------------------------------------------------
// GRU scan (H=8, IN=4, 10 classes), B=8192 rows, S=2048 steps, scalar NLL out.
// 512 independent single-wave workgroups, 16 batch rows per wave.
// Per step: 11x V_WMMA_F32_16X16X4_F32 (r,z,n gates + logits), one LDS
// transpose of h, per-lane 10-class log-softmax. x_batch is double-buffered
// into LDS with global_load_async_to_lds_b128 (ASYNCcnt), one chunk ahead.
// ---------------------------------------------------------------------------

typedef __attribute__((ext_vector_type(2))) float v2f;
typedef __attribute__((ext_vector_type(8))) float v8f;

#define HD    8      // hidden dim
#define IND   4      // input bits
#define NC    10     // output classes
#define BATCH 8192
#define SEQ   2048
#define CH    64     // steps per async x-chunk
#define NCHUNK (SEQ / CH)

static __device__ __forceinline__ v8f wmma4(v2f a, v2f b, v8f c) {
  // D = A(16x4) * B(4x16) + C(16x16), fp32.
  return __builtin_amdgcn_wmma_f32_16x16x4_f32(
      false, a, false, b, (short)0, c, false, false);
}

static __device__ __forceinline__ float fast_rcp(float x) {
  return __builtin_amdgcn_rcpf(x);           // single v_rcp_f32
}
static __device__ __forceinline__ float fast_sigmoid(float x) {
  return fast_rcp(1.0f + __expf(-x));
}
static __device__ __forceinline__ float fast_tanh(float x) {
  // 1 - 2/(e^{2x}+1); saturates correctly at +-1 for large |x|.
  return 1.0f - 2.0f * fast_rcp(__expf(2.0f * x) + 1.0f);
}

// Issue one 16x64-int chunk (4KB) of x rows into LDS: 8 async b128 transfers.
// Instruction t moves rows 2t+half; each lane carries 4 ints (16B).
static __device__ __forceinline__ void issue_chunk(const int* __restrict__ xbase,
                                                   int chunk, unsigned lds_base,
                                                   int half, int n) {
  const int j4 = n * 4;                      // int offset within row
#pragma unroll
  for (int t = 0; t < 8; ++t) {
    const int row = 2 * t + half;
    const int* g = xbase + (size_t)row * SEQ + chunk * CH + j4;
    const unsigned l = lds_base + (unsigned)(row * CH + j4) * 4u;
    asm volatile("global_load_async_to_lds_b128 %0, %1, off"
                 :: "v"(l), "v"((unsigned long long)(uintptr_t)g)
                 : "memory");
  }
}

static __device__ __forceinline__ void wait_async0() {
  asm volatile("s_wait_asynccnt 0" ::: "memory");
}

__global__ __launch_bounds__(32) void gru_scan_kernel(
    const int*   __restrict__ x_batch,
    const float* __restrict__ Wir, const float* __restrict__ bir,
    const float* __restrict__ Wiz, const float* __restrict__ biz,
    const float* __restrict__ Win, const float* __restrict__ bin_,
    const float* __restrict__ Whr, const float* __restrict__ bhr,
    const float* __restrict__ Whz, const float* __restrict__ bhz,
    const float* __restrict__ Whn, const float* __restrict__ bhn,
    const float* __restrict__ Wout, const float* __restrict__ bout,
    float* __restrict__ loss_acc)
{
  __shared__ float hbuf[16 * 16];        // h transpose staging
  __shared__ float lbuf[16 * 16];        // logits transpose staging
  __shared__ int   xstage[2][16][CH];    // double-buffered x chunks (8KB)

  const int lane = (int)threadIdx.x;      // 0..31 (wave32)
  const int half = lane >> 4;             // 0 | 1
  const int n    = lane & 15;             // column owned in B/C/D layouts
  const int m    = lane & 15;             // row owned in A layout
  const int k0   = 2 * half;              // A/B K index for vector element 0
  const int b0   = (int)blockIdx.x * 16;  // batch-tile origin

  // ---- B-layout weight fragments: element j holds Btile[k0+j][n] ----------
  v2f bWir = {}, bWiz = {}, bWin = {};
  v2f bWhr0 = {}, bWhr1 = {}, bWhz0 = {}, bWhz1 = {}, bWhn0 = {}, bWhn1 = {};
  v2f bWo0 = {}, bWo1 = {};
  if (n < HD) {
    bWir.x = Wir[n * IND + k0];     bWir.y = Wir[n * IND + k0 + 1];
    bWiz.x = Wiz[n * IND + k0];     bWiz.y = Wiz[n * IND + k0 + 1];
    bWin.x = Win[n * IND + k0];     bWin.y = Win[n * IND + k0 + 1];
    bWhr0.x = Whr[n * HD + k0];     bWhr0.y = Whr[n * HD + k0 + 1];
    bWhr1.x = Whr[n * HD + 4 + k0]; bWhr1.y = Whr[n * HD + 4 + k0 + 1];
    bWhz0.x = Whz[n * HD + k0];     bWhz0.y = Whz[n * HD + k0 + 1];
    bWhz1.x = Whz[n * HD + 4 + k0]; bWhz1.y = Whz[n * HD + 4 + k0 + 1];
    bWhn0.x = Whn[n * HD + k0];     bWhn0.y = Whn[n * HD + k0 + 1];
    bWhn1.x = Whn[n * HD + 4 + k0]; bWhn1.y = Whn[n * HD + 4 + k0 + 1];
  }
  if (n < NC) {
    bWo0.x = Wout[n * HD + k0];     bWo0.y = Wout[n * HD + k0 + 1];
    bWo1.x = Wout[n * HD + 4 + k0]; bWo1.y = Wout[n * HD + 4 + k0 + 1];
  }

  // ---- bias accumulator inits (C layout: same value in all 8 VGPRs) -------
  const float vr  = (n < HD) ? (bir[n] + bhr[n]) : 0.0f;
  const float vz  = (n < HD) ? (biz[n] + bhz[n]) : 0.0f;
  const float vhn = (n < HD) ? bhn[n]            : 0.0f;
  const float vxn = (n < HD) ? bin_[n]           : 0.0f;
  const float vbo = (n < NC) ? bout[n]           : 0.0f;
  v8f cr, cz, chn, cxn, cbo;
#pragma unroll
  for (int i = 0; i < 8; ++i) { cr[i]=vr; cz[i]=vz; chn[i]=vhn; cxn[i]=vxn; cbo[i]=vbo; }

  // ---- recurrent state ----------------------------------------------------
  v8f h_d = {};                 // h in C/D layout (lane owns column n, 8 rows)
  v2f ah0 = {}, ah1 = {};       // h in A layout (lane owns row m, K chunks)
  int   prev = 0;               // teacher-forced input count for row m
  float acc  = 0.0f;            // local NLL sum

  const int* __restrict__ xbase = x_batch + (size_t)b0 * SEQ;
  const unsigned lbase0 = (unsigned)(uintptr_t)&xstage[0][0][0];
  const unsigned lbase1 = (unsigned)(uintptr_t)&xstage[1][0][0];

  // prime chunk 0
  issue_chunk(xbase, 0, lbase0, half, n);
  wait_async0();

  for (int c = 0; c < NCHUNK; ++c) {
    // overlap: fetch chunk c+1 while computing chunk c
    if (c + 1 < NCHUNK)
      issue_chunk(xbase, c + 1, ((c + 1) & 1) ? lbase1 : lbase0, half, n);
    const int buf = c & 1;

    for (int sl = 0; sl < CH; ++sl) {
      // A-layout binarized input: x[:,k] = bit (IND-1-k) of prev (MSB first)
      v2f ax;
      ax.x = (float)((prev >> (IND - 1 - (k0 + 0))) & 1);
      ax.y = (float)((prev >> (IND - 1 - (k0 + 1))) & 1);

      // gate pre-activations: 9 WMMAs (K split 4 | 4+4)
      v8f pre_r = cr;
      pre_r = wmma4(ax,  bWir,  pre_r);
      pre_r = wmma4(ah0, bWhr0, pre_r);
      pre_r = wmma4(ah1, bWhr1, pre_r);
      v8f pre_z = cz;
      pre_z = wmma4(ax,  bWiz,  pre_z);
      pre_z = wmma4(ah0, bWhz0, pre_z);
      pre_z = wmma4(ah1, bWhz1, pre_z);
      v8f hn = chn;                 // h @ Whn.T + bhn  (gated by r below)
      hn = wmma4(ah0, bWhn0, hn);
      hn = wmma4(ah1, bWhn1, hn);
      v8f xn = cxn;                 // x @ Win.T + bin_
      xn = wmma4(ax, bWin, xn);

      // elementwise GRU update (D layout; padded columns n>=8 stay exactly 0)
#pragma unroll
      for (int i = 0; i < 8; ++i) {
        float r  = fast_sigmoid(pre_r[i]);
        float z  = fast_sigmoid(pre_z[i]);
        float nn = fast_tanh(xn[i] + r * hn[i]);
        h_d[i]   = nn + z * (h_d[i] - nn);   // (1-z)*nn + z*h
      }

      // transpose h: D layout -> LDS -> A layout (feeds logits + next step)
#pragma unroll
      for (int i = 0; i < 8; ++i)
        hbuf[(i + 8 * half) * 16 + n] = h_d[i];
      __syncthreads();
      ah0 = *(const v2f*)&hbuf[m * 16 + k0];
      ah1 = *(const v2f*)&hbuf[m * 16 + 4 + k0];

      // logits = h_next @ Wout.T + bout : 2 WMMAs
      v8f lg = cbo;
      lg = wmma4(ah0, bWo0, lg);
      lg = wmma4(ah1, bWo1, lg);
#pragma unroll
      for (int i = 0; i < 8; ++i)
        lbuf[(i + 8 * half) * 16 + n] = lg[i];

      const int cur = xstage[buf][m][sl];   // target @ s; prev for s+1 (LDS hit)
      __syncthreads();

      // per-lane 10-class log-softmax + NLL gather (lanes 0..15 own one row)
      if (lane < 16) {
        float mx = -3.0e38f;
#pragma unroll
        for (int j = 0; j < NC; ++j) mx = fmaxf(mx, lbuf[m * 16 + j]);
        float se = 0.0f;
#pragma unroll
        for (int j = 0; j < NC; ++j) se += __expf(lbuf[m * 16 + j] - mx);
        const float lse = mx + __logf(se);
        acc += lse - lbuf[m * 16 + cur];
      }
      __syncthreads();
      prev = cur;
    }

    wait_async0();   // chunk c+1 resident before next outer iteration
  }

  // wave reduction + one global atomic per wave
#pragma unroll
  for (int off = 16; off > 0; off >>= 1)
    acc += __shfl_down(acc, off, 32);
  if (lane == 0)
    atomicAdd(loss_acc, acc);
}

__global__ void zero_kernel(float* p) { p[0] = 0.0f; }

__global__ void finalize_kernel(const float* p, float* out) {
  out[0] = p[0] / (float)(BATCH * SEQ);
}

extern "C" void kernel_launch(void* const* d_in, const int* in_sizes, int n_in,
                              void* d_out, int out_size, void* d_ws, size_t ws_size,
                              hipStream_t stream) {
  const int*   x_batch = (const int*)  d_in[0];
  const float* Wir  = (const float*)d_in[1];  const float* bir  = (const float*)d_in[2];
  const float* Wiz  = (const float*)d_in[3];  const float* biz  = (const float*)d_in[4];
  const float* Win  = (const float*)d_in[5];  const float* bin_ = (const float*)d_in[6];
  const float* Whr  = (const float*)d_in[7];  const float* bhr  = (const float*)d_in[8];
  const float* Whz  = (const float*)d_in[9];  const float* bhz  = (const float*)d_in[10];
  const float* Whn  = (const float*)d_in[11]; const float* bhn  = (const float*)d_in[12];
  const float* Wout = (const float*)d_in[13]; const float* bout = (const float*)d_in[14];
  float* out = (float*)d_out;
  float* ws  = (float*)d_ws;

  zero_kernel<<<1, 1, 0, stream>>>(ws);
  gru_scan_kernel<<<BATCH / 16, 32, 0, stream>>>(
      x_batch, Wir, bir, Wiz, biz, Win, bin_, Whr, bhr, Whz, bhz, Whn, bhn,
      Wout, bout, ws);
  finalize_kernel<<<1, 1, 0, stream>>>(ws, out);
}